// MessagePassingNet_27943057228185
// MI455X (gfx1250) — compile-verified
//
#include <hip/hip_runtime.h>
#include <hip/hip_bf16.h>

typedef __attribute__((ext_vector_type(2))) float v2f;
typedef __attribute__((ext_vector_type(8))) float v8f;

// ---------------- init: zero output accumulator + degree counters ----------
__global__ void gcn_init_kernel(float* __restrict__ out, int* __restrict__ deg,
                                int n_out, int n_nodes) {
    int tid = blockIdx.x * blockDim.x + threadIdx.x;
    if (tid < n_out)   out[tid] = 0.0f;
    if (tid < n_nodes) deg[tid] = 0;
}

// ---------------- degree count over targets (col) --------------------------
__global__ void gcn_degree_kernel(const int* __restrict__ col, int* __restrict__ deg,
                                  int n_edges) {
    int e = blockIdx.x * blockDim.x + threadIdx.x;
    if (e < n_edges) atomicAdd(&deg[col[e]], 1);
}

// ---------------- dinv = rsqrt(deg + 1)  (+1 = self loop) ------------------
__global__ void gcn_dinv_kernel(const int* __restrict__ deg, float* __restrict__ dinv,
                                int n_nodes) {
    int i = blockIdx.x * blockDim.x + threadIdx.x;
    if (i < n_nodes) dinv[i] = __frsqrt_rn((float)(deg[i] + 1));
}

// ---------------- h = x @ W^T via V_WMMA_F32_16X16X4_F32 -------------------
// Block = 128 threads = 4 waves. Wave w computes the 16x16 tile
// h[m0:m0+16, 16w:16w+16].  K = 64 consumed as 16 WMMA steps of K=4.
//
// A (16x4, MxK) lane layout: lanes 0-15 -> rows M=lane, VGPR0/1 = K=k0,k0+1;
// lanes 16-31 -> rows M=lane-16, VGPR0/1 = K=k0+2,k0+3.
// B (4x16, KxN) mirrored: lane indexes N, VGPR indexes K.
// Since B[k][n] = W[n][k], both A and B use identical per-lane b64 gathers.
__global__ void __launch_bounds__(128)
gcn_gemm_wmma_kernel(const float* __restrict__ x, const float* __restrict__ W,
                     float* __restrict__ h) {
    const int wave = threadIdx.x >> 5;      // 0..3 -> N tile
    const int lane = threadIdx.x & 31;
    const int half = lane >> 4;             // 0: lanes 0-15, 1: lanes 16-31
    const int r    = lane & 15;
    const int m0   = blockIdx.x * 16;
    const int n0   = wave * 16;
    const int koff = half * 2;              // K sub-offset 0 or 2

    const float* __restrict__ xrow = x + (size_t)(m0 + r) * 64;
    const float* __restrict__ wrow = W + (size_t)(n0 + r) * 64;

    v8f c = {};
#pragma unroll
    for (int k0 = 0; k0 < 64; k0 += 4) {
        v2f a = *(const v2f*)(xrow + k0 + koff);   // b64 load: x[m][k0+koff..+1]
        v2f b = *(const v2f*)(wrow + k0 + koff);   // b64 load: W[n][k0+koff..+1]
        c = __builtin_amdgcn_wmma_f32_16x16x4_f32(
                /*neg_a=*/false, a, /*neg_b=*/false, b,
                /*c_mod=*/(short)0, c, /*reuse_a=*/false, /*reuse_b=*/false);
    }

    // C/D layout: lanes 0-15: VGPR v -> (M=m0+v, N=n0+lane);
    //             lanes 16-31: VGPR v -> (M=m0+v+8, N=n0+lane-16)
    const int mbase = m0 + half * 8;
    const int n     = n0 + r;
#pragma unroll
    for (int v = 0; v < 8; ++v) {
        h[(size_t)(mbase + v) * 64 + n] = c[v];
    }
}

// ---------------- edge scatter: out[col] += h[row] * dinv[r]*dinv[c] -------
// 16 threads per edge, float4 gather + 4 float atomics (L2-resident).
__global__ void gcn_scatter_kernel(const int* __restrict__ row,
                                   const int* __restrict__ col,
                                   const float* __restrict__ dinv,
                                   const float* __restrict__ h,
                                   float* __restrict__ out, int n_edges) {
    long long tid = (long long)blockIdx.x * blockDim.x + threadIdx.x;
    int e  = (int)(tid >> 4);
    if (e >= n_edges) return;
    int d4 = ((int)tid & 15) << 2;

    int r = row[e];
    int c = col[e];
    float nm = dinv[r] * dinv[c];

    const float4 hv = *(const float4*)(h + (size_t)r * 64 + d4);
    float* o = out + (size_t)c * 64 + d4;
    atomicAdd(o + 0, hv.x * nm);
    atomicAdd(o + 1, hv.y * nm);
    atomicAdd(o + 2, hv.z * nm);
    atomicAdd(o + 3, hv.w * nm);
}

// ---------------- finalize: + self-loop + bias, PReLU ----------------------
__global__ void gcn_finalize_kernel(float* __restrict__ out,
                                    const float* __restrict__ h,
                                    const float* __restrict__ dinv,
                                    const float* __restrict__ bias,
                                    const float* __restrict__ prelu_a,
                                    int n_nodes) {
    int tid = blockIdx.x * blockDim.x + threadIdx.x;
    if (tid >= n_nodes * 64) return;
    int i = tid >> 6;
    int d = tid & 63;
    float di = dinv[i];
    float v  = out[tid] + h[tid] * di * di + bias[d];
    float a  = prelu_a[0];
    out[tid] = (v >= 0.0f) ? v : a * v;
}

extern "C" void kernel_launch(void* const* d_in, const int* in_sizes, int n_in,
                              void* d_out, int out_size, void* d_ws, size_t ws_size,
                              hipStream_t stream) {
    const float* x         = (const float*)d_in[0];   // [N, 64]
    const int*   edge_idx  = (const int*)d_in[1];     // [2, E] flat
    const float* W         = (const float*)d_in[2];   // [64, 64]
    const float* bias      = (const float*)d_in[3];   // [64]
    const float* prelu_a   = (const float*)d_in[4];   // [1]
    float* out = (float*)d_out;                       // [N, 64]

    const int N = in_sizes[0] / 64;
    const int E = in_sizes[1] / 2;
    const int* row = edge_idx;
    const int* col = edge_idx + E;

    // Workspace layout: h [N*64 f32] | deg [N i32] | dinv [N f32]
    char* ws = (char*)d_ws;
    float* h    = (float*)ws;
    int*   deg  = (int*)(ws + (size_t)N * 64 * sizeof(float));
    float* dinv = (float*)((char*)deg + (size_t)N * sizeof(int));

    const int BT = 256;

    // 1) zero out-accumulator and degree counters
    {
        int total = N * 64;
        gcn_init_kernel<<<(total + BT - 1) / BT, BT, 0, stream>>>(out, deg, total, N);
    }
    // 2) degree over targets
    gcn_degree_kernel<<<(E + BT - 1) / BT, BT, 0, stream>>>(col, deg, E);
    // 3) dinv
    gcn_dinv_kernel<<<(N + BT - 1) / BT, BT, 0, stream>>>(deg, dinv, N);
    // 4) h = x @ W^T  (WMMA; 4 waves/block -> 16 rows x full 64 cols per block)
    gcn_gemm_wmma_kernel<<<N / 16, 128, 0, stream>>>(x, W, h);
    // 5) edge scatter (16 threads/edge)
    {
        long long threads = (long long)E * 16;
        int blocks = (int)((threads + BT - 1) / BT);
        gcn_scatter_kernel<<<blocks, BT, 0, stream>>>(row, col, dinv, h, out, E);
    }
    // 6) self-loop + bias + PReLU
    {
        int total = N * 64;
        gcn_finalize_kernel<<<(total + BT - 1) / BT, BT, 0, stream>>>(out, h, dinv, bias, prelu_a, N);
    }
}